// GCNConv_12154757447817
// MI455X (gfx1250) — compile-verified
//
#include <hip/hip_runtime.h>

typedef __attribute__((ext_vector_type(2))) float v2f;
typedef __attribute__((ext_vector_type(8))) float v8f;

#define F 128          // feature dim (fixed by reference)
#define LDS_STRIDE 136 // 128 + 8 pad: 2*136 mod 64 = 16 -> half-waves hit disjoint banks

// ---------------------------------------------------------------------------
// Native fp32 atomic add (CDNA5 L2 atomic unit, RNE, no-flush). Inline asm so
// we are guaranteed global_atomic_add_f32 and not a cmpswap loop.
// ---------------------------------------------------------------------------
__device__ __forceinline__ void atomic_add_f32(float* p, float v) {
    asm volatile("global_atomic_add_f32 %0, %1, off scope:SCOPE_DEV"
                 :: "v"(p), "v"(v) : "memory");
}

// ---------------------------------------------------------------------------
// Kernel 1: zero the accumulator (z lives in d_out)
// ---------------------------------------------------------------------------
__global__ void zero_kernel(float4* __restrict__ z4, int n4) {
    int i = blockIdx.x * blockDim.x + threadIdx.x;
    if (i < n4) z4[i] = make_float4(0.f, 0.f, 0.f, 0.f);
}

// ---------------------------------------------------------------------------
// Kernel 2: one wave per edge. Lane l handles features [4l, 4l+4):
//   z[row[e]][f] += vals[e] * x[col[e]][f]
// 512B coalesced b128 gather of x-row per wave; 4 f32 fadd atomics per lane.
// x and z both fit in the 192MB L2, so this runs at L2 rates.
// ---------------------------------------------------------------------------
__global__ void spmm_scatter_kernel(const int* __restrict__ row,
                                    const int* __restrict__ col,
                                    const float* __restrict__ vals,
                                    const float* __restrict__ x,
                                    float* z, int E) {
    int gtid = blockIdx.x * blockDim.x + threadIdx.x;
    int e    = gtid >> 5;
    int lane = threadIdx.x & 31;
    if (e >= E) return;

    int   r = row[e];
    int   c = col[e];
    float v = vals[e];

    const float4* xr = (const float4*)(x + (long)c * F);
    float4 xv = xr[lane];                      // global_load_b128, coalesced

    float* zp = z + (long)r * F + lane * 4;
    atomic_add_f32(zp + 0, v * xv.x);
    atomic_add_f32(zp + 1, v * xv.y);
    atomic_add_f32(zp + 2, v * xv.z);
    atomic_add_f32(zp + 3, v * xv.w);
}

// ---------------------------------------------------------------------------
// Kernel 3: in-place GEMM  out = z @ W  using V_WMMA_F32_16X16X4_F32.
// Block = 128 threads (4 waves). W (128x128 f32) staged in LDS once per block.
// Each wave owns one 16-row M-tile and produces all 128 output columns
// (8 accumulators of v8f), reusing each A-frag across 8 WMMAs.
// In-place is safe: a tile's rows are read and written only by its own wave,
// and all loads feed WMMAs that precede the stores.
// ---------------------------------------------------------------------------
__global__ void gemm_wmma_kernel(float* zo,                 // z in, out in place
                                 const float* __restrict__ w,
                                 int mtiles) {
    __shared__ float lw[F * LDS_STRIDE];

    // cooperative load of W into LDS (float4, coalesced)
    const float4* w4 = (const float4*)w;
    for (int i = threadIdx.x; i < (F * F) / 4; i += blockDim.x) {
        int r  = i >> 5;      // 32 float4 per 128-wide row
        int c4 = i & 31;
        float4 t = w4[i];
        float* dst = &lw[r * LDS_STRIDE + c4 * 4];
        dst[0] = t.x; dst[1] = t.y; dst[2] = t.z; dst[3] = t.w;
    }
    __syncthreads();

    int wave = threadIdx.x >> 5;
    int mt   = blockIdx.x * 4 + wave;
    if (mt >= mtiles) return;                  // wave-uniform: EXEC stays all-1s

    int lane = threadIdx.x & 31;
    int n    = lane & 15;                      // B/C/D column within tile
    int hi   = lane >> 4;                      // lane half
    int kb   = hi * 2;                         // A/B K sub-offset for this half

    const float* zrow = zo + (long)(mt * 16 + n) * F;  // A: m = lane&15 (== n)

    v8f acc[8] = {};

    for (int k = 0; k < F; k += 4) {
        // A-frag: 16x4 f32 -> lane holds z[m][k+kb], z[m][k+kb+1] (contiguous)
        v2f a = *(const v2f*)(zrow + k + kb);
#pragma unroll
        for (int nt = 0; nt < 8; nt++) {
            // B-frag: 4x16 f32 -> lane holds W[k+kb][nt*16+n], W[k+kb+1][...]
            v2f b;
            b.x = lw[(k + kb) * LDS_STRIDE + nt * 16 + n];
            b.y = lw[(k + kb + 1) * LDS_STRIDE + nt * 16 + n];
            acc[nt] = __builtin_amdgcn_wmma_f32_16x16x4_f32(
                /*neg_a=*/false, a, /*neg_b=*/false, b,
                /*c_mod=*/(short)0, acc[nt],
                /*reuse_a=*/false, /*reuse_b=*/false);
        }
    }

    // D layout: VGPR j -> row m = hi*8 + j, column n
#pragma unroll
    for (int nt = 0; nt < 8; nt++) {
#pragma unroll
        for (int j = 0; j < 8; j++) {
            int mrow = hi * 8 + j;
            zo[(long)(mt * 16 + mrow) * F + nt * 16 + n] = acc[nt][j];
        }
    }
}

// ---------------------------------------------------------------------------
// Launch: zero(d_out) -> edge scatter into d_out -> WMMA GEMM in place.
// ---------------------------------------------------------------------------
extern "C" void kernel_launch(void* const* d_in, const int* in_sizes, int n_in,
                              void* d_out, int out_size, void* d_ws, size_t ws_size,
                              hipStream_t stream) {
    const int*   row  = (const int*)  d_in[0];
    const int*   col  = (const int*)  d_in[1];
    const float* vals = (const float*)d_in[2];
    const float* x    = (const float*)d_in[3];
    const float* w    = (const float*)d_in[4];
    float*       out  = (float*)d_out;

    const int E = in_sizes[0];
    const int N = in_sizes[3] / F;      // 100000
    const int mtiles = N / 16;          // 6250 (N divisible by 16)

    // 1) z = 0 (accumulate in d_out)
    int n4 = out_size / 4;
    zero_kernel<<<(n4 + 255) / 256, 256, 0, stream>>>((float4*)out, n4);

    // 2) scatter-add: one wave32 per edge
    int waves_per_block = 256 / 32;
    int blocks = (E + waves_per_block - 1) / waves_per_block;
    spmm_scatter_kernel<<<blocks, 256, 0, stream>>>(row, col, vals, x, out, E);

    // 3) out = z @ W, in place, WMMA fp32
    gemm_wmma_kernel<<<(mtiles + 3) / 4, 128, 0, stream>>>(out, w, mtiles);
}